// GNNModel_79336635892006
// MI455X (gfx1250) — compile-verified
//
#include <hip/hip_runtime.h>
#include <hip/hip_bf16.h>
#include <cstdint>

typedef __attribute__((ext_vector_type(2))) float v2f;
typedef __attribute__((ext_vector_type(8))) float v8f;

// ---------------------------------------------------------------- utilities
__global__ void fill_kernel(float* __restrict__ p, float val, long long n) {
  long long i = (long long)blockIdx.x * blockDim.x + threadIdx.x;
  long long stride = (long long)gridDim.x * blockDim.x;
  for (; i < n; i += stride) p[i] = val;
}

__global__ void deg_accum_kernel(const int* __restrict__ dst,
                                 float* __restrict__ deg, int E) {
  int e = blockIdx.x * blockDim.x + threadIdx.x;
  if (e < E) atomicAdd(&deg[dst[e]], 1.0f);
}

__global__ void rsqrt_kernel(float* __restrict__ dinv, int n) {
  int i = blockIdx.x * blockDim.x + threadIdx.x;
  if (i < n) dinv[i] = rsqrtf(dinv[i]);
}

// ---------------------------------------------------------------- WMMA GEMM
// OUT[N x H] = X[N x K] @ W[K x H] via V_WMMA_F32_16X16X4_F32 (wave32).
// One wave computes one 16x16 output tile. blockDim.x = 32*(H/16),
// gridDim.x = ceil(N/16). EXEC is all-1s at every WMMA (guards only at store).
//
// VGPR layouts per CDNA5 ISA 7.12.2:
//   A (16x4 f32):  lane = m + 16*hl holds K = 2*hl + {0,1} in {a.x, a.y}
//   B (4x16 f32):  vgpr j, lane half hl: row K = j + 2*hl, col N = lane&15
//   C/D (16x16):   vgpr r, lane: row M = r + 8*hl, col N = lane&15
template <int K, int H>
__global__ void gemm_wmma_kernel(const float* __restrict__ X,
                                 const float* __restrict__ W,
                                 float* __restrict__ OUT, int N) {
  const int wave = threadIdx.x >> 5;
  const int lane = threadIdx.x & 31;
  const int hl   = lane >> 4;        // half-wave: 0 or 1
  const int m    = lane & 15;
  const int row0 = blockIdx.x * 16;
  const int col0 = wave * 16;

  int arow = row0 + m;
  if (arow >= N) arow = N - 1;       // clamp reads on (possible) tail tile
  const float* xrow = X + (size_t)arow * K;

  v8f acc = {};
#pragma unroll
  for (int k = 0; k < K; k += 4) {
    const int ka = k + 2 * hl;
    v2f a;
    a.x = xrow[ka];
    a.y = xrow[ka + 1];
    const float* wp = W + (size_t)ka * H + col0 + m;
    v2f b;
    b.x = wp[0];
    b.y = wp[H];
    acc = __builtin_amdgcn_wmma_f32_16x16x4_f32(
        /*neg_a=*/false, a, /*neg_b=*/false, b,
        /*c_mod=*/(short)0, acc, /*reuse_a=*/false, /*reuse_b=*/false);
  }

#pragma unroll
  for (int r = 0; r < 8; ++r) {
    int row = row0 + r + 8 * hl;
    if (row < N) OUT[(size_t)row * H + col0 + m] = acc[r];
  }
}

// ------------------------------------------------------------- edge scatter
// agg[dst] += h[src] * dinv[src]*dinv[dst].  HF = H/4; one thread handles one
// (edge, float4-chunk). Gathers are float4 (global_load_b128); h fits in L2.
template <int HF>
__global__ void scatter_kernel(const int* __restrict__ src,
                               const int* __restrict__ dst,
                               const float* __restrict__ dinv,
                               const float* __restrict__ h,
                               float* __restrict__ agg, int E) {
  int tid = blockIdx.x * blockDim.x + threadIdx.x;
  if (tid >= E * HF) return;                  // <= 51.2M, fits in int
  int e = tid / HF;
  int c = tid - e * HF;
  int s = src[e];
  int d = dst[e];
  float norm = dinv[s] * dinv[d];
  const float4* hv = (const float4*)(h + (size_t)s * (HF * 4));
  float4 v = hv[c];
  float* a = agg + (size_t)d * (HF * 4) + (size_t)c * 4;
  atomicAdd(a + 0, v.x * norm);
  atomicAdd(a + 1, v.y * norm);
  atomicAdd(a + 2, v.z * norm);
  atomicAdd(a + 3, v.w * norm);
}

// ------------------------------------------------- self-loop + bias + ReLU
template <int H>
__global__ void combine_relu_kernel(float* __restrict__ agg,
                                    const float* __restrict__ h,
                                    const float* __restrict__ dinv,
                                    const float* __restrict__ bias, int N) {
  int idx = blockIdx.x * blockDim.x + threadIdx.x;
  if (idx >= N * H) return;
  int i = idx / H;
  int f = idx - i * H;
  float di = dinv[i];
  float v = agg[idx] + h[idx] * di * di + bias[f];
  agg[idx] = v > 0.0f ? v : 0.0f;
}

// ------------------------------------------- classifier + log_softmax (C=10)
__global__ void classify_kernel(const float* __restrict__ h,
                                const float* __restrict__ Wc,
                                const float* __restrict__ bc,
                                float* __restrict__ out, int N) {
  __shared__ float sW[32 * 10];
  __shared__ float sb[10];
  for (int i = threadIdx.x; i < 320; i += blockDim.x) sW[i] = Wc[i];
  if (threadIdx.x < 10) sb[threadIdx.x] = bc[threadIdx.x];
  __syncthreads();

  int node = blockIdx.x * blockDim.x + threadIdx.x;
  if (node >= N) return;

  const float* hr = h + (size_t)node * 32;
  float logits[10];
#pragma unroll
  for (int c = 0; c < 10; ++c) logits[c] = sb[c];
#pragma unroll
  for (int k = 0; k < 32; ++k) {
    float hv = hr[k];
#pragma unroll
    for (int c = 0; c < 10; ++c) logits[c] += hv * sW[k * 10 + c];
  }
  float mx = logits[0];
#pragma unroll
  for (int c = 1; c < 10; ++c) mx = fmaxf(mx, logits[c]);
  float ssum = 0.0f;
#pragma unroll
  for (int c = 0; c < 10; ++c) ssum += expf(logits[c] - mx);
  float lse = mx + logf(ssum);
  float* o = out + (size_t)node * 10;
#pragma unroll
  for (int c = 0; c < 10; ++c) o[c] = logits[c] - lse;
}

// ------------------------------------------------------------------- driver
extern "C" void kernel_launch(void* const* d_in, const int* in_sizes, int n_in,
                              void* d_out, int out_size, void* d_ws,
                              size_t ws_size, hipStream_t stream) {
  const float* x  = (const float*)d_in[0];
  const int*   ei = (const int*)d_in[1];
  const float* W1 = (const float*)d_in[2];
  const float* b1 = (const float*)d_in[3];
  const float* W2 = (const float*)d_in[4];
  const float* b2 = (const float*)d_in[5];
  const float* Wc = (const float*)d_in[6];
  const float* bc = (const float*)d_in[7];
  float* out = (float*)d_out;

  const int N = in_sizes[0] / 128;
  const int E = in_sizes[1] / 2;
  const int* src = ei;       // edge_index[0]
  const int* dst = ei + E;   // edge_index[1]

  // workspace layout (floats): dinv[N] | h1[N*64] | agg1[N*64]
  // after layer 1, the h1 region is reused as h2[N*32] | agg2[N*32].
  float* dinv = (float*)d_ws;
  float* h1   = dinv + N;
  float* agg1 = h1 + (size_t)N * 64;
  float* h2   = h1;
  float* agg2 = h1 + (size_t)N * 32;

  const int ntiles = (N + 15) / 16;

  // degree -> dinv
  fill_kernel<<<1024, 256, 0, stream>>>(dinv, 1.0f, (long long)N);
  deg_accum_kernel<<<(E + 255) / 256, 256, 0, stream>>>(dst, dinv, E);
  rsqrt_kernel<<<(N + 255) / 256, 256, 0, stream>>>(dinv, N);

  // layer 1: h1 = X @ W1 ; propagate ; relu(agg + h*dinv^2 + b1) -> agg1
  gemm_wmma_kernel<128, 64><<<ntiles, 128, 0, stream>>>(x, W1, h1, N);
  fill_kernel<<<2048, 256, 0, stream>>>(agg1, 0.0f, (long long)N * 64);
  scatter_kernel<16><<<(int)(((long long)E * 16 + 255) / 256), 256, 0, stream>>>(
      src, dst, dinv, h1, agg1, E);
  combine_relu_kernel<64><<<(int)(((long long)N * 64 + 255) / 256), 256, 0,
                            stream>>>(agg1, h1, dinv, b1, N);

  // layer 2: h2 = out1 @ W2 ; propagate ; relu -> agg2
  gemm_wmma_kernel<64, 32><<<ntiles, 64, 0, stream>>>(agg1, W2, h2, N);
  fill_kernel<<<2048, 256, 0, stream>>>(agg2, 0.0f, (long long)N * 32);
  scatter_kernel<8><<<(int)(((long long)E * 8 + 255) / 256), 256, 0, stream>>>(
      src, dst, dinv, h2, agg2, E);
  combine_relu_kernel<32><<<(int)(((long long)N * 32 + 255) / 256), 256, 0,
                            stream>>>(agg2, h2, dinv, b2, N);

  // classifier + log_softmax
  classify_kernel<<<(N + 255) / 256, 256, 0, stream>>>(agg2, Wc, bc, out, N);
}